// COVtorch_12747462934620
// MI455X (gfx1250) — compile-verified
//
#include <hip/hip_runtime.h>

typedef __attribute__((ext_vector_type(2))) float v2f;
typedef __attribute__((ext_vector_type(8))) float v8f;

#define B_    32
#define N_    20000
#define D_    64
#define OUT_  256
#define NQ    (N_ / 4)          // 5000 K-steps of 4 rows
#define WPB   8                 // waves per block
#define CHUNKS 10               // blockIdx.y extent
#define WPBATCH (WPB * CHUNKS)  // 80 waves cooperating per batch

// upper-triangular tile enumeration (i<=j) of the 4x4 tile grid
__device__ __constant__ int TI[10] = {0,0,0,0,1,1,1,2,2,3};
__device__ __constant__ int TJ[10] = {0,1,2,3,1,2,3,2,3,3};

__global__ __launch_bounds__(256) void zero_kernel(float* p, int n) {
    int i = blockIdx.x * 256 + threadIdx.x;
    if (i < n) p[i] = 0.0f;
}

// load 4 fragments (all 64 columns of rows nA,nB for this lane) — streaming, non-temporal
__device__ __forceinline__ void load_quad_p(const float* __restrict__ p, v2f fr[4]) {
#pragma unroll
    for (int j = 0; j < 4; ++j) {
        fr[j].x = __builtin_nontemporal_load(p + 16 * j);        // row nA
        fr[j].y = __builtin_nontemporal_load(p + 16 * j + D_);   // row nB
    }
}

// vector column-sum accumulation: one v_pk_add_f32 per fragment, no repacking
__device__ __forceinline__ void csum_add(const v2f fr[4], v2f csum[4]) {
#pragma unroll
    for (int j = 0; j < 4; ++j) csum[j] += fr[j];
}

// 10 upper-tri tile WMMAs; fragments serve as both A (16x4) and B (4x16) operands
__device__ __forceinline__ void wmma_all(const v2f fr[4], v8f acc[10]) {
    int t = 0;
#pragma unroll
    for (int i = 0; i < 4; ++i) {
#pragma unroll
        for (int j = i; j < 4; ++j) {
            acc[t] = __builtin_amdgcn_wmma_f32_16x16x4_f32(
                false, fr[i], false, fr[j], (short)0, acc[t], false, false);
            ++t;
        }
    }
}

// Gram + column-sum accumulation. grid = (32, CHUNKS), block = 256 (8 waves).
__global__ __launch_bounds__(256) void gram_kernel(const float* __restrict__ x,
                                                   float* __restrict__ covS,
                                                   float* __restrict__ colS) {
    __shared__ float sCov[10 * 16 * 16];   // 10 upper-tri 16x16 tiles
    __shared__ float sCol[64];

    const int tid  = threadIdx.x;
    const int b    = blockIdx.x;
    const int wave = tid >> 5;
    const int lane = tid & 31;
    const int half = lane >> 4;
    const int col  = lane & 15;
    const int wib  = blockIdx.y * WPB + wave;   // 0..79, wave id within batch

    for (int f = tid; f < 10 * 256; f += 256) sCov[f] = 0.0f;
    if (tid < 64) sCol[tid] = 0.0f;
    __syncthreads();

    const float* xb = x + (size_t)b * N_ * D_;

    v8f acc[10] = {};
    v2f csum[4] = {};

    // provably wave-uniform trip count -> scalar loop control
    int rem = __builtin_amdgcn_readfirstlane((NQ - wib + WPBATCH - 1) / WPBATCH); // 62 or 63
    const float*  p      = xb + (size_t)(4 * wib + 2 * half) * D_ + col;
    const size_t  stride = (size_t)WPBATCH * 4 * D_;   // elements between this wave's quads

    // ping-pong double buffering with a guard-free steady loop:
    // at rem>=3 both prefetches (cur+1, cur+2) are in range -> single reaching def per buffer.
    v2f A[4], Bf[4];
    load_quad_p(p, A); p += stride;
    while (rem >= 3) {
        load_quad_p(p, Bf); p += stride;     // prefetch B while computing A
        csum_add(A, csum);
        wmma_all(A, acc);
        load_quad_p(p, A);  p += stride;     // prefetch A while computing B (unconditional)
        csum_add(Bf, csum);
        wmma_all(Bf, acc);
        rem -= 2;
    }
    if (rem == 2) {                          // final full pair, no further prefetch
        load_quad_p(p, Bf);
        csum_add(A, csum);
        wmma_all(A, acc);
        csum_add(Bf, csum);
        wmma_all(Bf, acc);
    } else {                                 // rem == 1
        csum_add(A, csum);
        wmma_all(A, acc);
    }

    // reduce this wave's partials into LDS (float DS atomics)
#pragma unroll
    for (int j = 0; j < 4; ++j) atomicAdd(&sCol[16 * j + col], csum[j].x + csum[j].y);

    {
        int t = 0;
#pragma unroll
        for (int i = 0; i < 4; ++i) {
#pragma unroll
            for (int j = i; j < 4; ++j) {
#pragma unroll
                for (int r = 0; r < 8; ++r) {
                    const int m = r + 8 * half;  // C layout: VGPR r -> M=r (lanes 0-15), M=r+8 (lanes 16-31)
                    atomicAdd(&sCov[t * 256 + m * 16 + col], acc[t][r]);
                }
                ++t;
            }
        }
    }
    __syncthreads();

    // one global-atomic flush per workgroup (L2-resident 520KB region)
    float* covB = covS + (size_t)b * (D_ * D_);
    for (int f = tid; f < 10 * 256; f += 256) {
        const int t = f >> 8, m = (f >> 4) & 15, c = f & 15;
        const int d = 16 * TI[t] + m, e = 16 * TJ[t] + c;
        atomicAdd(&covB[d * D_ + e], sCov[f]);
    }
    if (tid < 64) atomicAdd(&colS[b * 64 + tid], sCol[tid]);
}

// Per-batch: cov = (S - s s^T / N)/(N-1); out = feat@W + bias; L2-normalize.
__global__ __launch_bounds__(256) void finalize_kernel(const float* __restrict__ covS,
                                                       const float* __restrict__ colS,
                                                       const float* __restrict__ W,
                                                       const float* __restrict__ bias,
                                                       float* __restrict__ out) {
    __shared__ float feat[D_ * D_];
    __shared__ float red[256];

    const int b = blockIdx.x, tid = threadIdx.x;
    const float invN   = 1.0f / (float)N_;
    const float invNm1 = 1.0f / (float)(N_ - 1);
    const float* S = covS + (size_t)b * (D_ * D_);
    const float* s = colS + b * 64;

    for (int f = tid; f < D_ * D_; f += 256) {
        const int d = f >> 6, e = f & 63;
        const float raw = ((d >> 4) <= (e >> 4)) ? S[f] : S[e * 64 + d]; // symmetrize upper-tri
        feat[f] = (raw - s[d] * s[e] * invN) * invNm1;
    }
    __syncthreads();

    float accv = bias[tid];
    const float* Wc = W + tid;   // coalesced: thread tid reads column tid
#pragma unroll 4
    for (int k = 0; k < D_ * D_; ++k)
        accv = fmaf(feat[k], Wc[(size_t)k * OUT_], accv);

    red[tid] = accv * accv;
    __syncthreads();
    for (int sft = 128; sft > 0; sft >>= 1) {
        if (tid < sft) red[tid] += red[tid + sft];
        __syncthreads();
    }
    const float nrm = fmaxf(sqrtf(red[0]), 1e-12f);
    out[b * OUT_ + tid] = accv / nrm;
}

extern "C" void kernel_launch(void* const* d_in, const int* in_sizes, int n_in,
                              void* d_out, int out_size, void* d_ws, size_t ws_size,
                              hipStream_t stream) {
    const float* x    = (const float*)d_in[0];   // [32, 20000, 64]
    const float* W    = (const float*)d_in[1];   // [4096, 256]
    const float* bias = (const float*)d_in[2];   // [256]
    float* out  = (float*)d_out;                 // [32, 256]
    float* covS = (float*)d_ws;                  // [32][64][64]
    float* colS = covS + B_ * D_ * D_;           // [32][64]

    const int zn = B_ * D_ * D_ + B_ * D_;       // 133120 floats
    zero_kernel<<<(zn + 255) / 256, 256, 0, stream>>>(covS, zn);

    dim3 g1(B_, CHUNKS);
    gram_kernel<<<g1, 256, 0, stream>>>(x, covS, colS);

    finalize_kernel<<<B_, 256, 0, stream>>>(covS, colS, W, bias, out);
}